// Warping_72481868087624
// MI455X (gfx1250) — compile-verified
//
#include <hip/hip_runtime.h>
#include <cstdint>

// ---------------------------------------------------------------------------
// Trilinear DDF warping (DeepReg Warping layer) for MI455X / gfx1250.
//   ddf:   (4, 128, 160, 160, 3) f32
//   image: (4, 128, 160, 160)    f32
//   out:   (4, 128, 160, 160)    f32
// Memory-bound (~262 MB streamed => ~11 us floor @ 23.3 TB/s). No matmul
// structure => no WMMA. CDNA5-specific paths used instead:
//   - async global->LDS (ASYNCcnt) staging of the ddf tile with TH=NT so the
//     157 MB streamed ddf does not evict the 52 MB image from the 192 MB L2
//   - s_wait_asynccnt split-counter wait (no barrier: each lane reads back
//     only the floats it staged itself)
//   - NT store for the write-once output
//   - wave32 (32,8) tiling: each wave spans 32 consecutive z => coalesced
//     ddf/out and z-local gathers; 8 y-rows per block reuse gathered lines
//   - pure 32-bit offset arithmetic (all buffers < 2^31 B) => SGPR-base +
//     32-bit voffset addressing, no 64-bit VALU chains.
// ---------------------------------------------------------------------------

#if defined(__has_builtin)
#if __has_builtin(__builtin_amdgcn_global_load_async_to_lds_b32)
#define HAVE_ASYNC_LDS 1
#endif
#endif
#ifndef HAVE_ASYNC_LDS
#define HAVE_ASYNC_LDS 0
#endif

#define B_  4
#define D1_ 128
#define D2_ 160
#define D3_ 160

// gfx12-family CPol: TH in bits [2:0]; TH_NT == 1 (non-temporal).
#define CPOL_TH_NT 1

// Builtin signature (from hipcc diagnostic): as(1) int* / as(3) int*.
typedef __attribute__((address_space(1))) int* gas1_ip;
typedef __attribute__((address_space(3))) int* gas3_ip;

__global__ __launch_bounds__(256) void warp_ddf_kernel(
    const float* __restrict__ ddf,
    const float* __restrict__ img,
    float* __restrict__ out)
{
    constexpr int F2 = D2_, F3 = D3_;

    const uint32_t tx  = threadIdx.x;        // 0..31 -> z within tile
    const uint32_t ty  = threadIdx.y;        // 0..7  -> y within tile
    const uint32_t z   = blockIdx.x * 32u + tx;
    const uint32_t y   = blockIdx.y * 8u + ty;
    const uint32_t bfx = blockIdx.z;         // b*F1 + x
    const uint32_t xx  = bfx % D1_;
    const uint32_t bb  = bfx / D1_;

    // Voxel index shared by out-store and ddf-load (ddf base = 3*o).
    const uint32_t o     = (bfx * F2 + y) * F3 + z;
    const uint32_t dbase = o * 3u;

    float ux, uy, uz;

#if HAVE_ASYNC_LDS
    // Stage this block's ddf tile (256 threads x 3 floats = 3072 B) into LDS
    // with async global->LDS b32 copies (TH=NT). Thread (ty,tx) copies its own
    // three floats; per wave each async op covers one coalesced 384 B row.
    __shared__ float tile[8 * 32 * 3];
    {
        const uint32_t tid = ty * 32u + tx;  // 0..255
        #pragma unroll
        for (uint32_t p = 0; p < 3; ++p) {
            gas1_ip g = (gas1_ip)(uintptr_t)(ddf + dbase + p);
            // generic LDS pointer low 32 bits == LDS byte offset on AMDGPU
            gas3_ip l = (gas3_ip)(uintptr_t)(uint32_t)(uintptr_t)(&tile[tid * 3u + p]);
            __builtin_amdgcn_global_load_async_to_lds_b32(g, l, 0, CPOL_TH_NT);
        }
#if defined(__has_builtin) && __has_builtin(__builtin_amdgcn_s_wait_asynccnt)
        __builtin_amdgcn_s_wait_asynccnt(0);
#else
        asm volatile("s_wait_asynccnt 0" ::: "memory");
#endif
        // No __syncthreads(): each lane reads exactly the LDS words its own
        // (now-complete) async copies wrote.
        ux = tile[tid * 3u + 0];
        uy = tile[tid * 3u + 1];
        uz = tile[tid * 3u + 2];
    }
#else
    {
        // Fallback: direct coalesced NT loads (ddf is streamed once).
        const float* dp = ddf + dbase;
        ux = __builtin_nontemporal_load(dp + 0);
        uy = __builtin_nontemporal_load(dp + 1);
        uz = __builtin_nontemporal_load(dp + 2);
    }
#endif

    // Warped sample coordinate, clamped to the image bounds (reference: clip).
    float cx = fminf(fmaxf((float)xx + ux, 0.0f), (float)(D1_ - 1));
    float cy = fminf(fmaxf((float)y  + uy, 0.0f), (float)(D2_ - 1));
    float cz = fminf(fmaxf((float)z  + uz, 0.0f), (float)(D3_ - 1));

    const float fx = floorf(cx), fy = floorf(cy), fz = floorf(cz);
    const float wx = cx - fx,  wy = cy - fy,  wz = cz - fz;

    const uint32_t ix0 = (uint32_t)(int)fx;
    const uint32_t iy0 = (uint32_t)(int)fy;
    const uint32_t iz0 = (uint32_t)(int)fz;
    const uint32_t ix1 = min(ix0 + 1u, (uint32_t)(D1_ - 1));
    const uint32_t iy1 = min(iy0 + 1u, (uint32_t)(D2_ - 1));
    const uint32_t iz1 = min(iz0 + 1u, (uint32_t)(D3_ - 1));

    // Uniform SGPR base + 32-bit per-lane offsets for the 8 gathers.
    const float* imb = img + bb * (uint32_t)(D1_ * D2_ * D3_);
    const uint32_t r00 = (ix0 * D2_ + iy0) * D3_;
    const uint32_t r01 = (ix0 * D2_ + iy1) * D3_;
    const uint32_t r10 = (ix1 * D2_ + iy0) * D3_;
    const uint32_t r11 = (ix1 * D2_ + iy1) * D3_;

    // Image fits in L2 (52 MB < 192 MB): default RT hint is what we want.
    const float v000 = imb[r00 + iz0];
    const float v001 = imb[r00 + iz1];
    const float v010 = imb[r01 + iz0];
    const float v011 = imb[r01 + iz1];
    const float v100 = imb[r10 + iz0];
    const float v101 = imb[r10 + iz1];
    const float v110 = imb[r11 + iz0];
    const float v111 = imb[r11 + iz1];

    const float w0x = 1.0f - wx, w0y = 1.0f - wy, w0z = 1.0f - wz;

    // Same accumulation order & grouping as the reference loop (dx,dy,dz).
    float acc;
    acc  = ((w0x * w0y) * w0z) * v000;
    acc += ((w0x * w0y) * wz ) * v001;
    acc += ((w0x * wy ) * w0z) * v010;
    acc += ((w0x * wy ) * wz ) * v011;
    acc += ((wx  * w0y) * w0z) * v100;
    acc += ((wx  * w0y) * wz ) * v101;
    acc += ((wx  * wy ) * w0z) * v110;
    acc += ((wx  * wy ) * wz ) * v111;

    // Output is write-once: non-temporal store, don't pollute L2.
    __builtin_nontemporal_store(acc, out + o);
}

extern "C" void kernel_launch(void* const* d_in, const int* in_sizes, int n_in,
                              void* d_out, int out_size, void* d_ws, size_t ws_size,
                              hipStream_t stream) {
    (void)in_sizes; (void)n_in; (void)d_ws; (void)ws_size; (void)out_size;
    const float* ddf = (const float*)d_in[0];   // (4,128,160,160,3)
    const float* img = (const float*)d_in[1];   // (4,128,160,160)
    float* out = (float*)d_out;                 // (4,128,160,160)

    dim3 block(32, 8, 1);                        // 8 wave32s per block
    dim3 grid(D3_ / 32, D2_ / 8, B_ * D1_);      // (5, 20, 512)
    warp_ddf_kernel<<<grid, block, 0, stream>>>(ddf, img, out);
}